// DiffeqSolver_4252017623680
// MI455X (gfx1250) — compile-verified
//
#include <hip/hip_runtime.h>

// Neural-ODE RK4 on MI455X (gfx1250).
// One 16-row batch tile per 64-thread workgroup: a PAIR of waves splits the
// N-dimension (HT padded 100->128 = 8 N-tiles, 4 per wave, symmetric -> no
// divergence). Weights in LDS (transposed, bank-pad strides), state h stays in
// registers (WMMA A-layout) for all 199 steps, layer outputs bounce through
// double-buffered LDS (D-layout -> A-layout), 3 pair-local barriers per RK4
// stage. fp32 WMMA 16x16x4 keeps reference numerics.
// LDS/WG = 158.7 KB -> exactly 2 WGs colocate per 320 KB WGP.

typedef float v2f __attribute__((ext_vector_type(2)));
typedef float v8f __attribute__((ext_vector_type(8)));

#define NB   4096
#define NH   64
#define NHT  100
#define HTP  128    // padded hidden (8 * 16)
#define NT   200

// K-strides: stride % 64 == 4 -> wave32 ds_load_b64 of B-operands hits 64 banks.
#define S1   68     // stride for K=64 rows
#define SK   132    // stride for K=128 rows

#define OFF_W1T 0                        // [128][S1]  W1T[n][k] = W1[k][n]
#define OFF_W2T (HTP*S1)                 // 8704   [128][SK]
#define OFF_W3T (OFF_W2T + HTP*SK)       // 25600  [ 64][SK]
#define OFF_B1  (OFF_W3T + NH*SK)        // 34048
#define OFF_B2  (OFF_B1 + HTP)           // 34176
#define OFF_B3  (OFF_B2 + HTP)           // 34304
#define OFF_Z   (OFF_B3 + NH)            // 34368
#define ZT_SZ   (16*SK)                  // 2112 (one z buffer)
#define KB_SZ   (16*S1)                  // 1088 (k bounce buffer)
#define NLDS    (OFF_Z + 2*ZT_SZ + KB_SZ) // 39680 floats = 158720 B

#define WMMA_F32(a, b, c) \
  __builtin_amdgcn_wmma_f32_16x16x4_f32(false, (a), false, (b), (short)0, (c), false, false)

static __device__ __forceinline__ float fast_tanh(float x) {
#if __has_builtin(__builtin_amdgcn_tanhf)
  return __builtin_amdgcn_tanhf(x);            // v_tanh_f32 (TRANS pipe)
#elif __has_builtin(__builtin_amdgcn_tanh_f32)
  return __builtin_amdgcn_tanh_f32(x);
#else
  float e = __expf(2.0f * x);                  // v_exp_f32 (TRANS pipe)
  return 1.0f - 2.0f * __builtin_amdgcn_rcpf(e + 1.0f);
#endif
}

static __device__ __forceinline__ v8f splat8(float x) {
  v8f r;
  #pragma unroll
  for (int j = 0; j < 8; ++j) r[j] = x;
  return r;
}

__global__ void __launch_bounds__(64) __attribute__((amdgpu_waves_per_eu(4)))
node_rk4(const float* __restrict__ h0, const float* __restrict__ tg,
         const float* __restrict__ W1, const float* __restrict__ b1,
         const float* __restrict__ W2, const float* __restrict__ b2,
         const float* __restrict__ W3, const float* __restrict__ b3,
         float* __restrict__ out) {
  __shared__ __align__(16) float lds[NLDS];
  const int tid = threadIdx.x;

  // ---- one-time: zero pads, load transposed weights + biases ----
  for (int i = tid; i < OFF_Z; i += 64) lds[i] = 0.0f;
  __syncthreads();
  for (int i = tid; i < NH * NHT; i += 64) {             // W1 [64][100]
    int k = i / NHT, n = i - k * NHT;
    lds[OFF_W1T + n * S1 + k] = W1[i];
  }
  for (int i = tid; i < NHT * NHT; i += 64) {            // W2 [100][100]
    int k = i / NHT, n = i - k * NHT;
    lds[OFF_W2T + n * SK + k] = W2[i];
  }
  for (int i = tid; i < NHT * NH; i += 64) {             // W3 [100][64]
    int k = i / NH, n = i - k * NH;
    lds[OFF_W3T + n * SK + k] = W3[i];
  }
  for (int i = tid; i < NHT; i += 64) { lds[OFF_B1 + i] = b1[i]; lds[OFF_B2 + i] = b2[i]; }
  if (tid < NH) lds[OFF_B3 + tid] = b3[tid];
  __syncthreads();

  const int sub  = tid >> 5;             // which half of the wave pair
  const int lane = tid & 31;
  const int lm  = lane & 15;
  const int cb  = (lane >> 4) << 1;      // A/B layout: lanes 16-31 hold K+2,K+3
  const int hw8 = (lane >> 4) << 3;      // D layout: lanes 16-31 hold rows M+8
  float* zA = &lds[OFF_Z];
  float* zB = zA + ZT_SZ;
  float* kb = zB + ZT_SZ;
  const int n0 = 4 * sub;                // my 4 N-tiles in layers 1-2
  const int m0 = 2 * sub;                // my 2 N-tiles in layer 3

  float bias1v[4], bias2v[4], bias3v[2];
  #pragma unroll
  for (int j = 0; j < 4; ++j) {
    bias1v[j] = lds[OFF_B1 + 16 * (n0 + j) + lm];
    bias2v[j] = lds[OFF_B2 + 16 * (n0 + j) + lm];
  }
  #pragma unroll
  for (int j = 0; j < 2; ++j) bias3v[j] = lds[OFF_B3 + 16 * (m0 + j) + lm];

  const int row = blockIdx.x * 16 + lm;
  const float* hin = h0 + (long)row * NH + cb;
  float* orow = out + (long)row * NT * NH + cb;

  v2f ha[16], hacc[16], kv[16];
  #pragma unroll
  for (int c = 0; c < 16; ++c) {
    ha[c] = *(const v2f*)(hin + 4 * c);
    kv[c].x = 0.0f; kv[c].y = 0.0f;
  }
  #pragma unroll
  for (int c = 0; c < 8; ++c) {          // t=0 output, columns split across pair
    int cc = c + 8 * sub;
    *(v2f*)(orow + 4 * cc) = ha[cc];
  }

  #pragma unroll 1
  for (int s = 0; s < NT - 1; ++s) {
    const float dt = tg[s + 1] - tg[s];
    #pragma unroll
    for (int c = 0; c < 16; ++c) hacc[c] = ha[c];

    #pragma unroll 1
    for (int stage = 0; stage < 4; ++stage) {
      const float ci = (stage == 0) ? 0.0f : ((stage == 3) ? dt : 0.5f * dt);
      const float ca = (stage == 0 || stage == 3) ? dt * (1.0f / 6.0f) : dt * (1.0f / 3.0f);

      // ---- layer 1: zA = tanh((h + ci*k) @ W1 + b1); K=64, my N-tiles n0..n0+3 ----
      v8f acc[4];
      #pragma unroll
      for (int j = 0; j < 4; ++j) acc[j] = splat8(bias1v[j]);
      #pragma unroll
      for (int c = 0; c < 16; ++c) {
        v2f a = ha[c] + ci * kv[c];
        #pragma unroll
        for (int j = 0; j < 4; ++j) {
          v2f b = *(const v2f*)&lds[OFF_W1T + (16 * (n0 + j) + lm) * S1 + 4 * c + cb];
          acc[j] = WMMA_F32(a, b, acc[j]);
        }
      }
      #pragma unroll
      for (int j = 0; j < 4; ++j)
        #pragma unroll
        for (int r = 0; r < 8; ++r)
          zA[(r + hw8) * SK + 16 * (n0 + j) + lm] = fast_tanh(acc[j][r]);
      __syncthreads();

      // ---- layer 2: zB = tanh(zA @ W2 + b2); K=128 ----
      #pragma unroll
      for (int j = 0; j < 4; ++j) acc[j] = splat8(bias2v[j]);
      #pragma unroll
      for (int c = 0; c < 32; ++c) {
        v2f a = *(const v2f*)&zA[lm * SK + 4 * c + cb];
        #pragma unroll
        for (int j = 0; j < 4; ++j) {
          v2f b = *(const v2f*)&lds[OFF_W2T + (16 * (n0 + j) + lm) * SK + 4 * c + cb];
          acc[j] = WMMA_F32(a, b, acc[j]);
        }
      }
      #pragma unroll
      for (int j = 0; j < 4; ++j)
        #pragma unroll
        for (int r = 0; r < 8; ++r)
          zB[(r + hw8) * SK + 16 * (n0 + j) + lm] = fast_tanh(acc[j][r]);
      __syncthreads();

      // ---- layer 3: kb = zB @ W3 + b3; K=128, my N-tiles m0..m0+1 ----
      v8f a3[2];
      #pragma unroll
      for (int j = 0; j < 2; ++j) a3[j] = splat8(bias3v[j]);
      #pragma unroll
      for (int c = 0; c < 32; ++c) {
        v2f a = *(const v2f*)&zB[lm * SK + 4 * c + cb];
        #pragma unroll
        for (int j = 0; j < 2; ++j) {
          v2f b = *(const v2f*)&lds[OFF_W3T + (16 * (m0 + j) + lm) * SK + 4 * c + cb];
          a3[j] = WMMA_F32(a, b, a3[j]);
        }
      }
      #pragma unroll
      for (int j = 0; j < 2; ++j)
        #pragma unroll
        for (int r = 0; r < 8; ++r)
          kb[(r + hw8) * S1 + 16 * (m0 + j) + lm] = a3[j][r];
      __syncthreads();

      // k: D-layout -> A-layout (full K for both waves), RK4 accumulate
      #pragma unroll
      for (int c = 0; c < 16; ++c) {
        kv[c] = *(const v2f*)&kb[lm * S1 + 4 * c + cb];
        hacc[c] += ca * kv[c];
      }
    }

    #pragma unroll
    for (int c = 0; c < 16; ++c) ha[c] = hacc[c];
    #pragma unroll
    for (int c = 0; c < 8; ++c) {
      int cc = c + 8 * sub;
      *(v2f*)(orow + (long)(s + 1) * NH + 4 * cc) = ha[cc];
    }
  }
}

extern "C" void kernel_launch(void* const* d_in, const int* in_sizes, int n_in,
                              void* d_out, int out_size, void* d_ws, size_t ws_size,
                              hipStream_t stream) {
  const float* h0 = (const float*)d_in[0];
  const float* tg = (const float*)d_in[1];
  const float* W1 = (const float*)d_in[2];
  const float* b1 = (const float*)d_in[3];
  const float* W2 = (const float*)d_in[4];
  const float* b2 = (const float*)d_in[5];
  const float* W3 = (const float*)d_in[6];
  const float* b3 = (const float*)d_in[7];
  float* out = (float*)d_out;

  dim3 grid(NB / 16);   // 256 workgroups; one 16-row tile = 2 cooperating waves each
  dim3 block(64);
  hipLaunchKernelGGL(node_rk4, grid, block, 0, stream,
                     h0, tg, W1, b1, W2, b2, W3, b3, out);
}